// AttentionPooling_42485816492305
// MI455X (gfx1250) — compile-verified
//
#include <hip/hip_runtime.h>

typedef __attribute__((ext_vector_type(2))) float v2f;
typedef __attribute__((ext_vector_type(8))) float v8f;

#define E_DIM 768
#define D_DIM 512
#define H_DIM 8
#define B_DIM 128
#define WT_STRIDE 772  // padded row stride (772 % 64 == 4) -> 8 rows on distinct banks

// ---- CDNA5 async global->LDS copy (ASYNCcnt-tracked, bypasses VGPRs) ----
__device__ __forceinline__ void async_load_b128(unsigned lds_addr, unsigned long long gaddr) {
  asm volatile("global_load_async_to_lds_b128 %0, %1, off" ::"v"(lds_addr), "v"(gaddr)
               : "memory");
}
__device__ __forceinline__ void wait_async0() {
  asm volatile("s_wait_asynccnt 0x0" ::: "memory");
}

// ---------------------------------------------------------------- phase A1: q
__global__ __launch_bounds__(256) void k_qproj(const float* __restrict__ qry,
                                               const float* __restrict__ q_w,
                                               const float* __restrict__ in_b,
                                               float* __restrict__ q_out) {
  int d = blockIdx.x * blockDim.x + threadIdx.x;
  if (d < D_DIM) {
    float acc = 0.f;
    for (int j = 0; j < D_DIM; ++j) acc += qry[j] * q_w[d * D_DIM + j];
    q_out[d] = (acc + in_b[d]) * 0.125f;  // DH^-0.5 = 1/8
  }
}

// ------------------------------------------- phase A2: Wt[8][768] = K_h^T q_h
__global__ __launch_bounds__(256) void k_wt(const float* __restrict__ k_w,
                                            const float* __restrict__ q,
                                            float* __restrict__ wt) {
  int tid = blockIdx.x * blockDim.x + threadIdx.x;  // 8*768
  if (tid >= H_DIM * E_DIM) return;
  int row = tid / E_DIM;
  int e = tid % E_DIM;
  float acc = 0.f;
  for (int i = 0; i < 64; ++i) acc += k_w[(size_t)(row * 64 + i) * E_DIM + e] * q[row * 64 + i];
  wt[row * E_DIM + e] = acc;
}

// ------------------------------------------ phase B: scores = feats @ W (WMMA f32)
__global__ __launch_bounds__(256) void k_scores(const float* __restrict__ feats,
                                                const float* __restrict__ wt,
                                                float* __restrict__ scores, int N) {
  __shared__ float wtlds[H_DIM * WT_STRIDE];  // ~24.1KB
  __shared__ float tileA[2][128][36];         // 2 x 18KB, padded stride vs bank conflicts
  const int n0 = blockIdx.x * 128;
  const int tid = threadIdx.x;
  const int lane = tid & 31;
  const int wave = tid >> 5;
  const int lrow = lane & 15;
  const int koff = (lane < 16) ? 0 : 2;

  // stage wt (8x768) into LDS once per workgroup, coalesced
  for (int i = tid; i < H_DIM * E_DIM; i += 256) {
    wtlds[(i / E_DIM) * WT_STRIDE + (i % E_DIM)] = wt[i];
  }

  // async-staging addresses: 8 threads cover 32 cols (float4 each) of 4 row groups
  const int cbase = (tid & 7) * 4;
  const int rb = tid >> 3;
  unsigned ldsa[2][4];
  unsigned long long ga[4];
  for (int it = 0; it < 4; ++it) {
    int r = rb + it * 32;
    ldsa[0][it] = (unsigned)(uintptr_t)&tileA[0][r][cbase];
    ldsa[1][it] = (unsigned)(uintptr_t)&tileA[1][r][cbase];
    long gr = (long)n0 + r;
    if (gr > N - 1) gr = N - 1;
    ga[it] = (unsigned long long)(uintptr_t)(feats + gr * E_DIM + cbase);
  }
  // prologue: stage 0 -> buffer 0
  for (int it = 0; it < 4; ++it) async_load_b128(ldsa[0][it], ga[it]);

  v8f c = {0.f, 0.f, 0.f, 0.f, 0.f, 0.f, 0.f, 0.f};
  // lanes with lrow>=8 feed C columns 8..15 (never stored) -> just mirror row lrow&7
  const float* wbase = &wtlds[(lrow & 7) * WT_STRIDE + koff];

  for (int s = 0; s < 24; ++s) {  // 24 stages of 32 E-columns
    const int cur = s & 1;
    wait_async0();    // my stage-s tile landed in LDS
    __syncthreads();  // everyone's tile landed; everyone done reading other buffer
    if (s < 23) {
      unsigned long long boff = (unsigned long long)(s + 1) * 128ull;  // 32 floats
      for (int it = 0; it < 4; ++it) async_load_b128(ldsa[cur ^ 1][it], ga[it] + boff);
    }
    const float* arow = &tileA[cur][wave * 16 + lrow][koff];
    const float* brow = wbase + s * 32;
    for (int k = 0; k < 8; ++k) {
      v2f a = *reinterpret_cast<const v2f*>(arow + k * 4);
      v2f b = *reinterpret_cast<const v2f*>(brow + k * 4);
      c = __builtin_amdgcn_wmma_f32_16x16x4_f32(false, a, false, b, (short)0, c, false, false);
    }
  }
  if (lrow < H_DIM) {
    int mbase = n0 + wave * 16 + ((lane < 16) ? 0 : 8);
    for (int i = 0; i < 8; ++i) {
      int node = mbase + i;
      if (node < N) scores[(size_t)node * H_DIM + lrow] = c[i];
    }
  }
}

// ---------------------- phase C: per-graph softmax + attention-weighted pooling
__global__ __launch_bounds__(256) void k_pool(const float* __restrict__ feats,
                                              const float* __restrict__ scores,
                                              const int* __restrict__ ptr,
                                              float* __restrict__ pooled) {
  const int b = blockIdx.x;
  const long base = ptr[b];
  const int M = ptr[b + 1] - (int)base;  // 32..512
  __shared__ float probs[512][8];  // 16KB
  __shared__ float red[256][8];    // 8KB
  __shared__ float invd[8];
  const int t = threadIdx.x;

  float s0[8], s1[8], mx[8];
  for (int h = 0; h < 8; ++h) mx[h] = -INFINITY;
  const bool v0 = t < M, v1 = (t + 256) < M;
  if (v0)
    for (int h = 0; h < 8; ++h) { s0[h] = scores[(base + t) * 8 + h]; mx[h] = fmaxf(mx[h], s0[h]); }
  if (v1)
    for (int h = 0; h < 8; ++h) { s1[h] = scores[(base + t + 256) * 8 + h]; mx[h] = fmaxf(mx[h], s1[h]); }
  for (int h = 0; h < 8; ++h) red[t][h] = mx[h];
  __syncthreads();
  for (int off = 128; off > 0; off >>= 1) {
    if (t < off)
      for (int h = 0; h < 8; ++h) red[t][h] = fmaxf(red[t][h], red[t + off][h]);
    __syncthreads();
  }
  for (int h = 0; h < 8; ++h) mx[h] = red[0][h];
  __syncthreads();

  float sm[8];
  for (int h = 0; h < 8; ++h) sm[h] = 0.f;
  if (v0)
    for (int h = 0; h < 8; ++h) { float p = __expf(s0[h] - mx[h]); probs[t][h] = p; sm[h] += p; }
  if (v1)
    for (int h = 0; h < 8; ++h) { float p = __expf(s1[h] - mx[h]); probs[t + 256][h] = p; sm[h] += p; }
  for (int h = 0; h < 8; ++h) red[t][h] = sm[h];
  __syncthreads();
  for (int off = 128; off > 0; off >>= 1) {
    if (t < off)
      for (int h = 0; h < 8; ++h) red[t][h] += red[t + off][h];
    __syncthreads();
  }
  if (t < 8) invd[t] = 1.0f / red[0][t];
  __syncthreads();

  // pooled[h][b][e] = sum_m attn[m,h] * feat[m,e]; thread t owns e = t, t+256, t+512
  float a0[8], a1[8], a2[8];
  for (int h = 0; h < 8; ++h) { a0[h] = 0.f; a1[h] = 0.f; a2[h] = 0.f; }
  for (int m = 0; m < M; ++m) {
    const float* fr = &feats[(base + m) * E_DIM];
    float f0 = fr[t], f1 = fr[t + 256], f2 = fr[t + 512];
    float4 p0 = *reinterpret_cast<const float4*>(&probs[m][0]);  // LDS broadcast
    float4 p1 = *reinterpret_cast<const float4*>(&probs[m][4]);
    a0[0] += p0.x * f0; a1[0] += p0.x * f1; a2[0] += p0.x * f2;
    a0[1] += p0.y * f0; a1[1] += p0.y * f1; a2[1] += p0.y * f2;
    a0[2] += p0.z * f0; a1[2] += p0.z * f1; a2[2] += p0.z * f2;
    a0[3] += p0.w * f0; a1[3] += p0.w * f1; a2[3] += p0.w * f2;
    a0[4] += p1.x * f0; a1[4] += p1.x * f1; a2[4] += p1.x * f2;
    a0[5] += p1.y * f0; a1[5] += p1.y * f1; a2[5] += p1.y * f2;
    a0[6] += p1.z * f0; a1[6] += p1.z * f1; a2[6] += p1.z * f2;
    a0[7] += p1.w * f0; a1[7] += p1.w * f1; a2[7] += p1.w * f2;
  }
  for (int h = 0; h < 8; ++h) {
    float iv = invd[h];
    size_t o = ((size_t)h * B_DIM + b) * E_DIM;
    pooled[o + t] = a0[h] * iv;
    pooled[o + t + 256] = a1[h] * iv;
    pooled[o + t + 512] = a2[h] * iv;
  }
}

// ------------------- phase D: o[b, h*64+n] = pooled_h @ v_w_h^T + bv (WMMA f32)
__global__ __launch_bounds__(256) void k_ogemm(const float* __restrict__ pooled,
                                               const float* __restrict__ v_w,
                                               const float* __restrict__ in_b,
                                               float* __restrict__ o) {
  int wv = blockIdx.x * 8 + (threadIdx.x >> 5);  // 256 waves
  int lane = threadIdx.x & 31;
  int h = wv >> 5;         // 8 heads
  int bt = (wv >> 2) & 7;  // 8 row tiles of b
  int nt = wv & 3;         // 4 col tiles of DH
  int lrow = lane & 15;
  int koff = (lane < 16) ? 0 : 2;
  v8f c = {0.f, 0.f, 0.f, 0.f, 0.f, 0.f, 0.f, 0.f};
  const float* arow = pooled + ((size_t)h * B_DIM + bt * 16 + lrow) * E_DIM;
  const float* brow = v_w + ((size_t)(h * 64 + nt * 16 + lrow)) * E_DIM;
  for (int k0 = 0; k0 < E_DIM; k0 += 4) {
    v2f a = *reinterpret_cast<const v2f*>(arow + k0 + koff);
    v2f b = *reinterpret_cast<const v2f*>(brow + k0 + koff);
    c = __builtin_amdgcn_wmma_f32_16x16x4_f32(false, a, false, b, (short)0, c, false, false);
  }
  int col = h * 64 + nt * 16 + lrow;
  float bias = in_b[2 * D_DIM + col];  // bv
  int rbase = bt * 16 + ((lane < 16) ? 0 : 8);
  for (int i = 0; i < 8; ++i) o[(size_t)(rbase + i) * D_DIM + col] = c[i] + bias;
}

// ------------------------- phase E: out = o @ out_w^T + out_b (WMMA f32)
__global__ __launch_bounds__(256) void k_out(const float* __restrict__ o,
                                             const float* __restrict__ out_w,
                                             const float* __restrict__ out_b,
                                             float* __restrict__ out) {
  int wv = blockIdx.x * 8 + (threadIdx.x >> 5);  // 256 waves = 8 row x 32 col tiles
  int lane = threadIdx.x & 31;
  int bt = wv >> 5;
  int nt = wv & 31;
  int lrow = lane & 15;
  int koff = (lane < 16) ? 0 : 2;
  v8f c = {0.f, 0.f, 0.f, 0.f, 0.f, 0.f, 0.f, 0.f};
  const float* arow = o + (size_t)(bt * 16 + lrow) * D_DIM;
  const float* brow = out_w + (size_t)(nt * 16 + lrow) * D_DIM;
  for (int k0 = 0; k0 < D_DIM; k0 += 4) {
    v2f a = *reinterpret_cast<const v2f*>(arow + k0 + koff);
    v2f b = *reinterpret_cast<const v2f*>(brow + k0 + koff);
    c = __builtin_amdgcn_wmma_f32_16x16x4_f32(false, a, false, b, (short)0, c, false, false);
  }
  int col = nt * 16 + lrow;
  float bias = out_b[col];
  int rbase = bt * 16 + ((lane < 16) ? 0 : 8);
  for (int i = 0; i < 8; ++i) out[(size_t)(rbase + i) * D_DIM + col] = c[i] + bias;
}

extern "C" void kernel_launch(void* const* d_in, const int* in_sizes, int n_in,
                              void* d_out, int out_size, void* d_ws, size_t ws_size,
                              hipStream_t stream) {
  const float* graph_emb = (const float*)d_in[0];
  const float* qry       = (const float*)d_in[1];
  const float* q_w       = (const float*)d_in[2];
  const float* k_w       = (const float*)d_in[3];
  const float* v_w       = (const float*)d_in[4];
  const float* in_b      = (const float*)d_in[5];
  const float* out_w     = (const float*)d_in[6];
  const float* out_b     = (const float*)d_in[7];
  const int*   ptr       = (const int*)d_in[8];
  (void)n_in; (void)out_size; (void)ws_size;

  const int N = in_sizes[0] / E_DIM;

  float* ws     = (float*)d_ws;
  float* q      = ws;                       // 512
  float* wt     = q + D_DIM;                // 8*768
  float* scores = wt + H_DIM * E_DIM;       // N*8
  float* pooled = scores + (size_t)N * 8;   // 8*128*768
  float* o      = pooled + (size_t)H_DIM * B_DIM * E_DIM;  // 128*512

  k_qproj<<<2, 256, 0, stream>>>(qry, q_w, in_b, q);
  k_wt<<<24, 256, 0, stream>>>(k_w, q, wt);
  int nb = (N + 127) / 128;
  k_scores<<<nb, 256, 0, stream>>>(graph_emb, wt, scores, N);
  k_pool<<<B_DIM, 256, 0, stream>>>(graph_emb, scores, ptr, pooled);
  k_ogemm<<<32, 256, 0, stream>>>(pooled, v_w, in_b, o);
  k_out<<<32, 256, 0, stream>>>(o, out_w, out_b, (float*)d_out);
}